// OverlappedSpatialReductionAttention_81106162418410
// MI455X (gfx1250) — compile-verified
//
#include <hip/hip_runtime.h>
#include <math.h>

// OverlappedSpatialReductionAttention for MI455X (gfx1250), fp32 end-to-end.
// All GEMMs use V_WMMA_F32_16X16X4_F32 (full-precision WMMA), since the
// problem is HBM-bound (~14 GFLOP vs ~110 MB of traffic at 23.3 TB/s).

typedef float v2f __attribute__((ext_vector_type(2)));
typedef float v8f __attribute__((ext_vector_type(8)));

#define WMMA_F32(a, b, c) \
  __builtin_amdgcn_wmma_f32_16x16x4_f32(false, (a), false, (b), (short)0, (c), false, false)

// ---------------------------------------------------------------------------
// Kernel 1: Q projection.  q[b,n,d] = sum_c x[b,n,c]*Wq[d,c] + bq[d]
// One wave -> one 16(n) x 16(d) tile; K-loop of 64 x (16x16x4) f32 WMMAs.
// A frag: A[m][k]=x[b,n0+m,c0+k] ; B frag: B[k][nn]=Wq[d0+nn,c0+k]
// Lane map: m/nn = lane&15, k = 2*(lane>>4) + vgpr  -> per-lane float2 loads.
// ---------------------------------------------------------------------------
__global__ __launch_bounds__(256) void k_qproj(const float* __restrict__ x,
                                               const float* __restrict__ Wq,
                                               const float* __restrict__ bq,
                                               float* __restrict__ q) {
  const int lane = threadIdx.x & 31;
  const int wid  = blockIdx.x * 8 + (threadIdx.x >> 5);   // 32768 tiles
  const int dt = wid & 15;            // 16 d-tiles
  const int nt = (wid >> 4) & 255;    // 256 n-tiles
  const int b  = wid >> 12;           // 8 batches
  const int m = lane & 15, g = lane >> 4;

  const float* arow = x  + ((size_t)(b * 4096 + nt * 16 + m)) * 256 + 2 * g;
  const float* brow = Wq + ((size_t)(dt * 16 + m)) * 256 + 2 * g;

  v8f acc = {};
#pragma unroll 8
  for (int kc = 0; kc < 256; kc += 4) {
    v2f a  = *(const v2f*)(arow + kc);
    v2f bb = *(const v2f*)(brow + kc);
    acc = WMMA_F32(a, bb, acc);
  }
  const float bias = bq[dt * 16 + m];
  float* orow = q + ((size_t)(b * 4096 + nt * 16)) * 256 + dt * 16 + m;
#pragma unroll
  for (int r = 0; r < 8; ++r)
    orow[(size_t)(r + 8 * g) * 256] = acc[r] + bias;
}

// ---------------------------------------------------------------------------
// Kernel 2: spatial reduction: dwconv7 s4 p3 + BN1 + GELU(exact) + ch-scale + BN2.
// Block = 256 threads (c = tid), block per (b, oy, ox) -> coalesced x loads.
// Output kv_pre laid out (b, s=oy*16+ox, c) so the KV GEMM reads contiguously.
// ---------------------------------------------------------------------------
__global__ __launch_bounds__(256) void k_sr(const float* __restrict__ x,
                                            const float* __restrict__ w7,
                                            const float* __restrict__ g1, const float* __restrict__ b1,
                                            const float* __restrict__ m1, const float* __restrict__ v1,
                                            const float* __restrict__ w2s,
                                            const float* __restrict__ g2, const float* __restrict__ b2,
                                            const float* __restrict__ m2, const float* __restrict__ v2,
                                            float* __restrict__ kv_pre) {
  const int c  = threadIdx.x;
  const int bs = blockIdx.x;          // b*256 + s
  const int b = bs >> 8, s = bs & 255;
  const int oy = s >> 4, ox = s & 15;

  float acc = 0.f;
  for (int ky = 0; ky < 7; ++ky) {
    const int iy = oy * 4 - 3 + ky;
    if (iy < 0 || iy > 63) continue;
    for (int kx = 0; kx < 7; ++kx) {
      const int ix = ox * 4 - 3 + kx;
      if (ix < 0 || ix > 63) continue;
      acc += w7[c * 49 + ky * 7 + kx] *
             x[((size_t)(b * 4096 + iy * 64 + ix)) * 256 + c];
    }
  }
  float sc1 = g1[c] * rsqrtf(v1[c] + 1e-5f);
  float y = acc * sc1 + (b1[c] - m1[c] * sc1);
  y = 0.5f * y * (1.0f + erff(y * 0.70710678118654752f));   // exact GELU
  y *= w2s[c];
  float sc2 = g2[c] * rsqrtf(v2[c] + 1e-5f);
  y = y * sc2 + (b2[c] - m2[c] * sc2);
  kv_pre[(size_t)bs * 256 + c] = y;
}

// ---------------------------------------------------------------------------
// Kernel 3: residual local 3x3 depthwise conv (pad 1):  kv = dw3(kv_pre)+lb+kv_pre
// ---------------------------------------------------------------------------
__global__ __launch_bounds__(256) void k_local(const float* __restrict__ kv_pre,
                                               const float* __restrict__ lw,
                                               const float* __restrict__ lb,
                                               float* __restrict__ kv) {
  const int c  = threadIdx.x;
  const int bs = blockIdx.x;
  const int b = bs >> 8, s = bs & 255;
  const int oy = s >> 4, ox = s & 15;

  float acc = kv_pre[(size_t)bs * 256 + c] + lb[c];
  for (int ky = 0; ky < 3; ++ky) {
    const int iy = oy + ky - 1;
    if (iy < 0 || iy > 15) continue;
    for (int kx = 0; kx < 3; ++kx) {
      const int ix = ox + kx - 1;
      if (ix < 0 || ix > 15) continue;
      acc += lw[c * 9 + ky * 3 + kx] *
             kv_pre[((size_t)(b * 256 + iy * 16 + ix)) * 256 + c];
    }
  }
  kv[(size_t)bs * 256 + c] = acc;
}

// ---------------------------------------------------------------------------
// Kernel 4: KV projection. kvp[b,d2,s] = sum_c Wkv[d2,c]*kv[b,s,c] + bkv[d2]
// Stored (b, 512, 256): rows 0..255 are K (head-major, hd), 256..511 are V^T.
// ---------------------------------------------------------------------------
__global__ __launch_bounds__(256) void k_kvproj(const float* __restrict__ kv,
                                                const float* __restrict__ Wkv,
                                                const float* __restrict__ bkv,
                                                float* __restrict__ kvp) {
  const int lane = threadIdx.x & 31;
  const int wid  = blockIdx.x * 8 + (threadIdx.x >> 5);   // 4096 tiles
  const int st = wid & 15;           // 16 s-tiles
  const int dt = (wid >> 4) & 31;    // 32 d2-tiles
  const int b  = wid >> 9;           // 8 batches
  const int m = lane & 15, g = lane >> 4;

  const float* arow = Wkv + ((size_t)(dt * 16 + m)) * 256 + 2 * g;
  const float* brow = kv  + ((size_t)(b * 256 + st * 16 + m)) * 256 + 2 * g;

  v8f acc = {};
#pragma unroll 8
  for (int kc = 0; kc < 256; kc += 4) {
    v2f a  = *(const v2f*)(arow + kc);
    v2f bb = *(const v2f*)(brow + kc);
    acc = WMMA_F32(a, bb, acc);
  }
#pragma unroll
  for (int r = 0; r < 8; ++r) {
    const int d2 = dt * 16 + r + 8 * g;
    kvp[((size_t)(b * 512 + d2)) * 256 + st * 16 + m] = acc[r] + bkv[d2];
  }
}

// ---------------------------------------------------------------------------
// Kernel 5: attention. Block = 64 threads (2 waves); each wave owns 16 query
// rows of one (b, head). Scores -> LDS (padded stride 260 to spread banks),
// wave-local softmax, then P·V with 2 output accumulators (hd = 32).
// ---------------------------------------------------------------------------
__global__ __launch_bounds__(64) void k_attn(const float* __restrict__ q,
                                             const float* __restrict__ kvp,
                                             float* __restrict__ out) {
  __shared__ float P[2][16][260];               // 33,280 B
  const int lane = threadIdx.x & 31;
  const int w    = threadIdx.x >> 5;            // wave in block: 0..1
  const int blk  = blockIdx.x;                  // 8192 blocks
  const int nblk = blk & 127;                   // 4096 / 32
  const int head = (blk >> 7) & 7;
  const int b    = blk >> 10;
  const int n0   = nblk * 32 + w * 16;
  const int m = lane & 15, g = lane >> 4;

  // Preload 8 Q A-fragments (reused for all 16 score tiles).
  const float* qb = q + ((size_t)(b * 4096 + n0 + m)) * 256 + head * 32 + 2 * g;
  v2f qa[8];
#pragma unroll
  for (int kk = 0; kk < 8; ++kk) qa[kk] = *(const v2f*)(qb + kk * 4);

  // S = (Q_h · K_h) * scale -> LDS.  K[i][s] = kvp[b, head*32+i, s]
  const float* Kbase = kvp + ((size_t)(b * 512 + head * 32)) * 256;
  for (int t = 0; t < 16; ++t) {
    v8f acc = {};
#pragma unroll
    for (int kk = 0; kk < 8; ++kk) {
      const int krow = kk * 4 + 2 * g;
      v2f bb;
      bb.x = Kbase[(size_t)krow * 256 + t * 16 + m];
      bb.y = Kbase[(size_t)(krow + 1) * 256 + t * 16 + m];
      acc = WMMA_F32(qa[kk], bb, acc);
    }
#pragma unroll
    for (int r = 0; r < 8; ++r)
      P[w][r + 8 * g][t * 16 + m] = acc[r] * 0.17677669529663688f;  // 32^-0.5
  }
  __syncthreads();

  // Row softmax: lane handles row (lane&15), half g covers 128 columns.
  {
    float* row = P[w][m];
    float mx = -3.4e38f;
    for (int j = 0; j < 128; ++j) mx = fmaxf(mx, row[g * 128 + j]);
    mx = fmaxf(mx, __shfl_xor(mx, 16, 32));
    float sum = 0.f;
    for (int j = 0; j < 128; ++j) {
      const float e = expf(row[g * 128 + j] - mx);
      row[g * 128 + j] = e;
      sum += e;
    }
    sum += __shfl_xor(sum, 16, 32);
    const float inv = 1.0f / sum;
    for (int j = 0; j < 128; ++j) row[g * 128 + j] *= inv;
  }
  __syncthreads();

  // O = P · V.  V^T[i][s] = kvp[b, 256 + head*32 + i, s]; B[k][i]=V[s=k][i].
  const float* Vbase = kvp + ((size_t)(b * 512 + 256 + head * 32)) * 256;
  v8f acc0 = {}, acc1 = {};
  for (int kc = 0; kc < 256; kc += 4) {
    v2f a  = *(const v2f*)(&P[w][m][kc + 2 * g]);
    v2f b0 = *(const v2f*)(Vbase + (size_t)m * 256 + kc + 2 * g);         // i = 0..15
    v2f b1 = *(const v2f*)(Vbase + (size_t)(16 + m) * 256 + kc + 2 * g);  // i = 16..31
    acc0 = WMMA_F32(a, b0, acc0);
    acc1 = WMMA_F32(a, b1, acc1);
  }
  float* obase = out + ((size_t)(b * 4096 + n0)) * 256 + head * 32;
#pragma unroll
  for (int r = 0; r < 8; ++r) {
    obase[(size_t)(r + 8 * g) * 256 + m]      = acc0[r];
    obase[(size_t)(r + 8 * g) * 256 + 16 + m] = acc1[r];
  }
}

// ---------------------------------------------------------------------------
extern "C" void kernel_launch(void* const* d_in, const int* in_sizes, int n_in,
                              void* d_out, int out_size, void* d_ws, size_t ws_size,
                              hipStream_t stream) {
  const float* x    = (const float*)d_in[0];
  // d_in[1], d_in[2]: h=64, w=64 (hardcoded)
  const float* Wq   = (const float*)d_in[3];
  const float* bq   = (const float*)d_in[4];
  const float* Wkv  = (const float*)d_in[5];
  const float* bkv  = (const float*)d_in[6];
  const float* w7   = (const float*)d_in[7];
  const float* g1   = (const float*)d_in[8];
  const float* b1   = (const float*)d_in[9];
  const float* m1   = (const float*)d_in[10];
  const float* v1   = (const float*)d_in[11];
  const float* w2s  = (const float*)d_in[12];
  const float* g2   = (const float*)d_in[13];
  const float* b2   = (const float*)d_in[14];
  const float* m2   = (const float*)d_in[15];
  const float* v2   = (const float*)d_in[16];
  const float* lw   = (const float*)d_in[17];
  const float* lb   = (const float*)d_in[18];
  float* out = (float*)d_out;

  // Workspace layout (floats): small buffers first, q last (33.5 MB).
  float* ws     = (float*)d_ws;
  float* kv_pre = ws;                         //  524,288
  float* kv     = kv_pre + 524288;            //  524,288
  float* kvp    = kv + 524288;                // 1,048,576
  float* qbuf   = kvp + 1048576;              // 8,388,608
  const size_t need = (size_t)(524288 * 2 + 1048576 + 8388608) * sizeof(float);
  // If scratch is tight, alias q onto d_out: each attention block reads
  // exactly the q region it later overwrites (read-before-write per block).
  float* q = (ws_size >= need) ? qbuf : out;

  k_qproj <<<4096, 256, 0, stream>>>(x, Wq, bq, q);
  k_sr    <<<2048, 256, 0, stream>>>(x, w7, g1, b1, m1, v1, w2s, g2, b2, m2, v2, kv_pre);
  k_local <<<2048, 256, 0, stream>>>(kv_pre, lw, lb, kv);
  k_kvproj<<< 512, 256, 0, stream>>>(kv, Wkv, bkv, kvp);
  k_attn  <<<8192,  64, 0, stream>>>(q, kvp, out);
}